// TaskAlignedAssigner_29240137351333
// MI455X (gfx1250) — compile-verified
//
#include <hip/hip_runtime.h>
#include <cstdint>

// Problem sizes from the reference (setup_inputs): B=32, A=33600, G=128, C=1
#define B_    32
#define A_    33600
#define G_    128
#define C_    1
#define TOPK_ 10
#define GTPB  4      // GTs processed per block in the top-k kernel (anchor reuse)
#define TPB   256    // 8 wave32 waves per block
#define CHUNK 512    // anchors staged per TDM chunk (8KB bbox + 2KB score)

typedef unsigned int u32x4_t __attribute__((ext_vector_type(4)));
typedef int          i32x4_t __attribute__((ext_vector_type(4)));
typedef int          i32x8_t __attribute__((ext_vector_type(8)));

// TDM 1-D load: DMA `tile_elems` f32 from global `ga` into LDS byte offset
// `lds_off`. `tensor_elems` = elements remaining from the tile start (OOB
// reads beyond it return zero). Descriptor per CDNA5 ISA ch.8 (D# groups).
__device__ __forceinline__ void tdm_load_1d(unsigned lds_off,
                                            unsigned long long ga,
                                            unsigned tile_elems,
                                            unsigned tensor_elems) {
#if defined(__HIP_DEVICE_COMPILE__)
  // group0: count=1 | lds_addr | global_addr[56:0] | type=2 ("image")
  u32x4_t g0v = {1u, lds_off, (unsigned)(ga & 0xFFFFFFFFu),
                 (unsigned)((ga >> 32) & 0x01FFFFFFu) | (2u << 30)};
  // group1: data_size=2 (4B), tensor_dim0=tensor_elems, tensor_dim1=1,
  //         tile_dim0=tile_elems, tile_dim1=1, stride0=tensor_elems
  i32x8_t g1v = {(int)0x00020000,
                 (int)((tensor_elems & 0xFFFFu) << 16),
                 (int)(((tensor_elems >> 16) & 0xFFFFu) | (1u << 16)),
                 (int)((tile_elems & 0xFFFFu) << 16),
                 1,
                 (int)tensor_elems,
                 (int)((tensor_elems & 0xFFFFu) << 16),
                 0};
  i32x4_t z4 = {0, 0, 0, 0};
#if defined(__clang_major__) && (__clang_major__ >= 23)
  i32x8_t z8 = {0, 0, 0, 0, 0, 0, 0, 0};
  __builtin_amdgcn_tensor_load_to_lds(g0v, g1v, z4, z4, z8, 0);
#else
  __builtin_amdgcn_tensor_load_to_lds(g0v, g1v, z4, z4, 0);
#endif
#endif
}

__device__ __forceinline__ unsigned lds_offset(const void* p) {
  // LDS aperture places the byte offset in flat addr[31:0].
  return (unsigned)(unsigned long long)(uintptr_t)p;
}

// ---------------------------------------------------------------- kernel 1
__global__ __launch_bounds__(TPB) void init_kernel(int* __restrict__ best,
                                                   int* __restrict__ best_gt) {
  int i = blockIdx.x * TPB + threadIdx.x;
  if (i < B_ * A_) {
    best[i]    = (int)0xFF800000;  // -inf as float bits
    best_gt[i] = G_;
  }
}

// ---------------------------------------------------------------- kernel 2
// One block handles (batch b, 4 consecutive GTs). Anchor bboxes+scores are
// streamed into LDS with double-buffered TDM DMA (TENSORcnt protocol); each
// thread keeps 4 sorted (val desc, idx asc) top-10 lists in registers, then a
// tournament merge in LDS produces the exact per-GT column top-10.
__global__ __launch_bounds__(TPB) void topk_kernel(
    const float* __restrict__ scores,    // (B, A, C) with C==1
    const float* __restrict__ bboxes,    // (B, A, 4)
    const float* __restrict__ gts,       // (B, G, 4)
    float* __restrict__ tkv,             // (B, G, TOPK)
    int*   __restrict__ tki) {           // (B, G, TOPK)
  __shared__ __align__(16) float s_bb[2][CHUNK * 4];  // staged bboxes
  __shared__ __align__(16) float s_sc[2][CHUNK];      // staged scores
  __shared__ float sv[TPB * TOPK_];
  __shared__ int   si[TPB * TOPK_];

  const int  tid   = threadIdx.x;
  const int  b     = blockIdx.y;
  const int  g0    = blockIdx.x * GTPB;
  const bool wave0 = (tid >> 5) == 0;    // uniform per wave

  float4 gbox[GTPB];
  float  garea[GTPB];
  bool   gvalid[GTPB];
#pragma unroll
  for (int gg = 0; gg < GTPB; ++gg) {
    const float* gp = gts + ((size_t)b * G_ + g0 + gg) * 4;
    gbox[gg]  = make_float4(gp[0], gp[1], gp[2], gp[3]);
    gvalid[gg] = (fabsf(gbox[gg].x) + fabsf(gbox[gg].y) +
                  fabsf(gbox[gg].z) + fabsf(gbox[gg].w)) > 0.0f;
    garea[gg] = (gbox[gg].z - gbox[gg].x) * (gbox[gg].w - gbox[gg].y);
  }

  float tv[GTPB][TOPK_];
  int   ti[GTPB][TOPK_];
#pragma unroll
  for (int gg = 0; gg < GTPB; ++gg)
#pragma unroll
    for (int k = 0; k < TOPK_; ++k) { tv[gg][k] = -INFINITY; ti[gg][k] = 0; }

  const unsigned long long bb_base =
      (unsigned long long)(uintptr_t)(bboxes + (size_t)b * A_ * 4);
  const unsigned long long sc_base =
      (unsigned long long)(uintptr_t)(scores + (size_t)b * A_);  // C == 1
  const int nchunk = (A_ + CHUNK - 1) / CHUNK;

  // Prologue: DMA chunk 0 into buffer 0.
  if (wave0) {
    tdm_load_1d(lds_offset(&s_bb[0][0]), bb_base, CHUNK * 4, (unsigned)A_ * 4);
    tdm_load_1d(lds_offset(&s_sc[0][0]), sc_base, CHUNK, (unsigned)A_);
  }

  for (int c = 0; c < nchunk; ++c) {
    const int base = c * CHUNK;
    if (wave0) {
      if (c + 1 < nchunk) {
        // Prefetch next chunk into the other buffer (freed by the barrier at
        // the end of the previous iteration), then wait for chunk c's two
        // in-order TDM ops: outstanding <= 2 means chunk c has landed.
        const int nb = base + CHUNK;
        tdm_load_1d(lds_offset(&s_bb[(c + 1) & 1][0]),
                    bb_base + (unsigned long long)nb * 16ull, CHUNK * 4,
                    (unsigned)(A_ - nb) * 4);
        tdm_load_1d(lds_offset(&s_sc[(c + 1) & 1][0]),
                    sc_base + (unsigned long long)nb * 4ull, CHUNK,
                    (unsigned)(A_ - nb));
#if defined(__HIP_DEVICE_COMPILE__)
        __builtin_amdgcn_s_wait_tensorcnt(2);
#endif
      } else {
#if defined(__HIP_DEVICE_COMPILE__)
        __builtin_amdgcn_s_wait_tensorcnt(0);
#endif
      }
    }
    __syncthreads();  // chunk c visible to all waves

    const float4* pb = (const float4*)&s_bb[c & 1][0];
    const float*  ps = &s_sc[c & 1][0];
#pragma unroll
    for (int half = 0; half < CHUNK / TPB; ++half) {
      const int k = tid + half * TPB;
      const int a = base + k;
      if (a < A_) {
        float4 p      = pb[k];
        float  salpha = sqrtf(ps[k]);              // score ** 0.5
        float  parea  = (p.z - p.x) * (p.w - p.y);
#pragma unroll
        for (int gg = 0; gg < GTPB; ++gg) {
          if (!gvalid[gg]) continue;
          float l  = fmaxf(p.x, gbox[gg].x);
          float t  = fmaxf(p.y, gbox[gg].y);
          float r  = fminf(p.z, gbox[gg].z);
          float bo = fminf(p.w, gbox[gg].w);
          float inter = fmaxf(r - l, 0.0f) * fmaxf(bo - t, 0.0f);
          float iou   = inter / (parea + garea[gg] - inter + 1e-6f);
          float i2    = iou * iou;
          float al    = salpha * (i2 * i2 * i2);   // s^0.5 * iou^6
          if (al > tv[gg][TOPK_ - 1]) {            // strict >: low idx wins ties
            tv[gg][TOPK_ - 1] = al;
            ti[gg][TOPK_ - 1] = a;
#pragma unroll
            for (int j = TOPK_ - 1; j > 0; --j) {  // bubble up (strict >)
              if (tv[gg][j] > tv[gg][j - 1]) {
                float fv = tv[gg][j]; tv[gg][j] = tv[gg][j - 1]; tv[gg][j - 1] = fv;
                int   fi = ti[gg][j]; ti[gg][j] = ti[gg][j - 1]; ti[gg][j - 1] = fi;
              }
            }
          }
        }
      }
    }
    __syncthreads();  // all waves done with buffer (c&1) before it is re-DMA'd
  }

  // Tournament merge of 256 sorted 10-lists per GT.
  for (int gg = 0; gg < GTPB; ++gg) {
#pragma unroll
    for (int k = 0; k < TOPK_; ++k) {
      sv[tid * TOPK_ + k] = tv[gg][k];
      si[tid * TOPK_ + k] = ti[gg][k];
    }
    __syncthreads();
    for (int stride = 1; stride < TPB; stride <<= 1) {
      if ((tid & (2 * stride - 1)) == 0) {
        const int ao  = tid * TOPK_;
        const int bo2 = (tid + stride) * TOPK_;
        int pa = 0, pb2 = 0;
        float ov[TOPK_]; int oi[TOPK_];
#pragma unroll
        for (int k = 0; k < TOPK_; ++k) {
          float va = sv[ao + pa], vb = sv[bo2 + pb2];
          int   ia = si[ao + pa], ib = si[bo2 + pb2];
          bool ta = (va > vb) || ((va == vb) && (ia < ib));
          ov[k] = ta ? va : vb;
          oi[k] = ta ? ia : ib;
          pa  += ta ? 1 : 0;
          pb2 += ta ? 0 : 1;
        }
#pragma unroll
        for (int k = 0; k < TOPK_; ++k) { sv[ao + k] = ov[k]; si[ao + k] = oi[k]; }
      }
      __syncthreads();
    }
    if (tid == 0) {
      const size_t base = ((size_t)b * G_ + g0 + gg) * TOPK_;
      for (int k = 0; k < TOPK_; ++k) { tkv[base + k] = sv[k]; tki[base + k] = si[k]; }
    }
    __syncthreads();
  }
}

// ---------------------------------------------------------------- kernel 3
// best[a] = max over finite top-k entries; non-negative floats compare
// correctly as signed-int bit patterns (init = -inf bits).
__global__ __launch_bounds__(TPB) void scatter_max_kernel(
    const float* __restrict__ tkv, const int* __restrict__ tki,
    int* __restrict__ best) {
  int t = blockIdx.x * TPB + threadIdx.x;
  if (t >= B_ * G_ * TOPK_) return;
  float v = tkv[t];
  if (v >= 0.0f) {  // finite entries are always >= 0; -inf means invalid GT
    int a = tki[t];
    int b = t / (G_ * TOPK_);
    atomicMax(&best[(size_t)b * A_ + a], __float_as_int(v));
  }
}

// ---------------------------------------------------------------- kernel 4
__global__ __launch_bounds__(TPB) void scatter_win_kernel(
    const float* __restrict__ tkv, const int* __restrict__ tki,
    const int* __restrict__ best, int* __restrict__ best_gt) {
  int t = blockIdx.x * TPB + threadIdx.x;
  if (t >= B_ * G_ * TOPK_) return;
  float v = tkv[t];
  if (v >= 0.0f) {
    int a = tki[t];
    int b = t / (G_ * TOPK_);
    int g = (t / TOPK_) % G_;
    if (__float_as_int(v) == best[(size_t)b * A_ + a])
      atomicMin(&best_gt[(size_t)b * A_ + a], g);
  }
}

// ---------------------------------------------------------------- kernel 5
// TDM: DMA the per-batch GT table (128 x 4 f32 = 2KB) into LDS, then gather.
__global__ __launch_bounds__(TPB) void finalize_kernel(
    const int*   __restrict__ best_gt,   // (B, A)
    const float* __restrict__ gts,       // (B, G, 4)
    const int*   __restrict__ gt_cls,    // (B, G)
    float*       __restrict__ out) {
  __shared__ __align__(16) float4 s_gts[G_];
  __shared__ int s_compact[G_];
  const int tid = threadIdx.x;
  const int b   = blockIdx.y;

#if defined(__HIP_DEVICE_COMPILE__)
  if ((tid >> 5) == 0) {  // wave 0 issues the TDM op (once per wave)
    tdm_load_1d(lds_offset(&s_gts[0]),
                (unsigned long long)(uintptr_t)(gts + (size_t)b * G_ * 4),
                G_ * 4, G_ * 4);
    __builtin_amdgcn_s_wait_tensorcnt(0);
  }
#else
  // Host-pass placeholder (never executed on device).
  for (int g = tid; g < G_; g += TPB) {
    const float* gp = gts + ((size_t)b * G_ + g) * 4;
    s_gts[g] = make_float4(gp[0], gp[1], gp[2], gp[3]);
  }
#endif
  __syncthreads();

  if (tid == 0) {  // compact = cumsum(valid) - 1  (128 steps, negligible)
    int run = 0;
    for (int g = 0; g < G_; ++g) {
      float4 q = s_gts[g];
      run += ((fabsf(q.x) + fabsf(q.y) + fabsf(q.z) + fabsf(q.w)) > 0.0f) ? 1 : 0;
      s_compact[g] = run - 1;
    }
  }
  __syncthreads();

  const int a = blockIdx.x * TPB + tid;
  if (a >= A_) return;
  const size_t ba = (size_t)b * A_ + a;
  const int  bg  = best_gt[ba];
  const bool pos = bg < G_;
  const int  bgs = pos ? bg : 0;
  float4 tb = pos ? s_gts[bgs] : make_float4(0.f, 0.f, 0.f, 0.f);
  float clsv = 0.0f;
  if (pos) {
    int lab = gt_cls[(size_t)b * G_ + bgs];  // one_hot(lab, C=1)
    clsv = (lab == 0) ? 1.0f : 0.0f;
  }
  const size_t BA = (size_t)B_ * A_;
  out[ba] = clsv;                                   // t_cls    (B,A,1)
  ((float4*)(out + BA))[ba] = tb;                   // t_bbox   (B,A,4)
  out[5 * BA + ba] = clsv;                          // t_scores (B,A,1)
  out[6 * BA + ba] = pos ? 1.0f : 0.0f;             // t_mask   (B,A)
  out[7 * BA + ba] = pos ? (float)s_compact[bgs] : -1.0f;  // t_gt_idx (B,A)
}

// ---------------------------------------------------------------- launch
extern "C" void kernel_launch(void* const* d_in, const int* in_sizes, int n_in,
                              void* d_out, int out_size, void* d_ws, size_t ws_size,
                              hipStream_t stream) {
  const float* scores    = (const float*)d_in[0];  // (B, A, C)
  const float* bboxes    = (const float*)d_in[1];  // (B, A, 4)
  // d_in[2] = anchors (A, 2) -- unused by the reference
  const int*   gt_cls    = (const int*)d_in[3];    // (B, G)
  const float* gt_bboxes = (const float*)d_in[4];  // (B, G, 4)
  float*       out       = (float*)d_out;

  const size_t BA = (size_t)B_ * A_;
  const int    NE = B_ * G_ * TOPK_;

  int*   best    = (int*)d_ws;                 // B*A  (float bits)
  int*   best_gt = best + BA;                  // B*A
  float* tkv     = (float*)(best_gt + BA);     // B*G*TOPK
  int*   tki     = (int*)(tkv + NE);           // B*G*TOPK

  init_kernel<<<(int)((BA + TPB - 1) / TPB), TPB, 0, stream>>>(best, best_gt);
  topk_kernel<<<dim3(G_ / GTPB, B_), TPB, 0, stream>>>(scores, bboxes, gt_bboxes,
                                                       tkv, tki);
  scatter_max_kernel<<<(NE + TPB - 1) / TPB, TPB, 0, stream>>>(tkv, tki, best);
  scatter_win_kernel<<<(NE + TPB - 1) / TPB, TPB, 0, stream>>>(tkv, tki, best,
                                                               best_gt);
  finalize_kernel<<<dim3((A_ + TPB - 1) / TPB, B_), TPB, 0, stream>>>(
      best_gt, gt_bboxes, gt_cls, out);
}